// Net_50440095924400
// MI455X (gfx1250) — compile-verified
//
#include <hip/hip_runtime.h>

// ---------------- problem constants (from reference) ----------------
#define LTREE 18
#define NN    262143          // 2^18 - 1 nodes
#define DD    128
#define EE    8
#define LEAF0 131071          // first leaf index = 2^17 - 1
#define NLEAF 131072
#define ALIGNR 32             // bucket alignment (rows per GEMM block)
#define MAXR  (NLEAF + 256)   // max padded rows per level (8 buckets * 31 pad)
#define LDA   132             // LDS A-tile stride (floats), avoids bank conflicts

typedef __attribute__((ext_vector_type(2))) float v2f;
typedef __attribute__((ext_vector_type(8))) float v8f;

#define WMMA_F32(a, b, c) \
  __builtin_amdgcn_wmma_f32_16x16x4_f32(false, (a), false, (b), (short)0, (c), false, false)

// ------------------------------------------------------------------
// Core: one 128-thread block computes C[32 x 128] = A(32x128, LDS) * B(128x128 row-major)
// via V_WMMA_F32_16X16X4_F32. Wave w owns columns [w*32, w*32+32); each B fragment
// is shared across both 16-row M-tiles (halves B-load issues per WMMA).
// A layout per ISA: lanes 0-15 row=lane, VGPR0/1 = K k0+0/k0+1; lanes 16-31 same rows, K k0+2/3.
// ------------------------------------------------------------------
__device__ __forceinline__ void wmma_32x128(const float* As, const float* __restrict__ B,
                                            v8f& c00, v8f& c01, v8f& c10, v8f& c11,
                                            int wv, int lane) {
  const int half = lane >> 4;
  const int lr   = lane & 15;
  const float* Arow0 = As + lr * LDA;          // M-tile 0 (rows 0..15)
  const float* Arow1 = As + (16 + lr) * LDA;   // M-tile 1 (rows 16..31)
  const int n0 = wv * 32;
  #pragma unroll 4
  for (int k0 = 0; k0 < DD; k0 += 4) {
    const int kb = k0 + half * 2;
    v2f a0, a1;
    a0.x = Arow0[kb];     a0.y = Arow0[kb + 1];
    a1.x = Arow1[kb];     a1.y = Arow1[kb + 1];
    v2f b0, b1;
    b0.x = B[(size_t)kb * DD + n0 + lr];
    b0.y = B[(size_t)(kb + 1) * DD + n0 + lr];
    b1.x = B[(size_t)kb * DD + n0 + 16 + lr];
    b1.y = B[(size_t)(kb + 1) * DD + n0 + 16 + lr];
    c00 = WMMA_F32(a0, b0, c00);
    c01 = WMMA_F32(a0, b1, c01);
    c10 = WMMA_F32(a1, b0, c10);
    c11 = WMMA_F32(a1, b1, c11);
  }
}

// ------------------------------------------------------------------
// Kernel 1: raw = data_vecs[data] @ data_W + data_b  (all N nodes, 32 rows/block)
// Also initializes ret = raw for leaf nodes.
// ------------------------------------------------------------------
__global__ __launch_bounds__(128) void init_gemm(
    const int* __restrict__ data, const float* __restrict__ data_vecs,
    const float* __restrict__ dataW, const float* __restrict__ data_b,
    float* __restrict__ raw, float* __restrict__ ret) {
  __shared__ float As[32 * LDA];
  __shared__ int srcIdx[32];
  const int t = threadIdx.x;
  const int row0 = blockIdx.x * 32;
  if (t < 32) {
    int node = row0 + t;
    srcIdx[t] = (node < NN) ? data[node] : 0;
  }
  __syncthreads();
  { // gather 32 rows of data_vecs into LDS: 4 threads/row, 8 float4 each
    const int r = t >> 2, q = t & 3;
    const float4* src = (const float4*)(data_vecs + (size_t)srcIdx[r] * DD) + q * 8;
    float4* dst = (float4*)(As + r * LDA) + q * 8;   // r*LDA*4 = 528*r bytes, 16B aligned
    #pragma unroll
    for (int i = 0; i < 8; ++i) dst[i] = src[i];
  }
  __syncthreads();
  const int wv = t >> 5, lane = t & 31;
  v8f c00 = {0.f,0.f,0.f,0.f,0.f,0.f,0.f,0.f}, c01 = c00, c10 = c00, c11 = c00;
  wmma_32x128(As, dataW, c00, c01, c10, c11, wv, lane);
  const int half = lane >> 4, lr = lane & 15;
  const int col = wv * 32 + lr;
  const float bias0 = data_b[col], bias1 = data_b[col + 16];
  #pragma unroll
  for (int v = 0; v < 8; ++v) {
    const int r = v + half * 8;
    // M-tile 0
    int node = row0 + r;
    if (node < NN) {
      float x0 = c00[v] + bias0, x1 = c01[v] + bias1;
      size_t o = (size_t)node * DD;
      raw[o + col] = x0;  raw[o + col + 16] = x1;
      if (node >= LEAF0) { ret[o + col] = x0; ret[o + col + 16] = x1; }
    }
    // M-tile 1
    node = row0 + 16 + r;
    if (node < NN) {
      float x0 = c10[v] + bias0, x1 = c11[v] + bias1;
      size_t o = (size_t)node * DD;
      raw[o + col] = x0;  raw[o + col + 16] = x1;
      if (node >= LEAF0) { ret[o + col] = x0; ret[o + col + 16] = x1; }
    }
  }
}

// ------------------------------------------------------------------
// Per-level bucketing machinery
// ------------------------------------------------------------------
__global__ void meta_init(int* __restrict__ cnt, int* __restrict__ cursor) {
  if (threadIdx.x < EE) { cnt[threadIdx.x] = 0; cursor[threadIdx.x] = 0; }
}

__global__ void level_hist(const int* __restrict__ edges, int c0, int M, int* __restrict__ cnt) {
  int j = blockIdx.x * 256 + threadIdx.x;
  if (j < M) atomicAdd(&cnt[edges[c0 + j]], 1);
}

// 1 block: 32-aligned bucket offsets, tile(32-row)->edge map, zero pad rows of Abuf
__global__ void level_offsets(const int* __restrict__ cnt, int* __restrict__ offs,
                              int* __restrict__ tileEdge, int ntiles,
                              float* __restrict__ Abuf) {
  __shared__ int so[EE], sc[EE];
  const int t = threadIdx.x;
  if (t == 0) {
    int o = 0;
    for (int e = 0; e < EE; ++e) {
      sc[e] = cnt[e];
      so[e] = o;
      offs[e] = o;
      o += (sc[e] + (ALIGNR - 1)) & ~(ALIGNR - 1);
    }
  }
  __syncthreads();
  for (int tt = t; tt < ntiles; tt += blockDim.x) {
    int e = -1;
    #pragma unroll
    for (int q = 0; q < EE; ++q) {
      int beg = so[q], end = so[q] + ((sc[q] + (ALIGNR - 1)) & ~(ALIGNR - 1));
      if (tt * ALIGNR >= beg && tt * ALIGNR < end) e = q;
    }
    tileEdge[tt] = e;
  }
  // zero padding rows so padded WMMA tiles contribute nothing
  for (int q = 0; q < EE; ++q) {
    int beg = so[q] + sc[q];
    int end = so[q] + ((sc[q] + (ALIGNR - 1)) & ~(ALIGNR - 1));
    int nel = (end - beg) * DD;
    for (int i = t; i < nel; i += blockDim.x) Abuf[(size_t)beg * DD + i] = 0.f;
  }
}

// assign each child a slot in its edge bucket; copy its ret row into Abuf[slot]
__global__ __launch_bounds__(256) void level_gather(
    const int* __restrict__ edges, const float* __restrict__ ret, int c0, int M,
    const int* __restrict__ offs, int* __restrict__ cursor,
    int* __restrict__ pos, float* __restrict__ Abuf) {
  __shared__ int slot_s[64];
  const int base = blockIdx.x * 64;
  const int t = threadIdx.x;
  if (t < 64) {
    int j = base + t;
    int s = -1;
    if (j < M) {
      int e = edges[c0 + j];
      s = offs[e] + atomicAdd(&cursor[e], 1);
      pos[j] = s;
    }
    slot_s[t] = s;
  }
  __syncthreads();
  const int r = t >> 2, q = t & 3;   // 4 threads per row, 8 float4 each
  const int j = base + r;
  if (j < M) {
    const int s = slot_s[r];
    const float4* src = (const float4*)(ret + (size_t)(c0 + j) * DD) + q * 8;
    float4* dst = (float4*)(Abuf + (size_t)s * DD) + q * 8;
    #pragma unroll
    for (int i = 0; i < 8; ++i) dst[i] = src[i];
  }
}

// ------------------------------------------------------------------
// Kernel: per 32-row tile (single edge id), C = A_tile @ edge_W[e]
// ------------------------------------------------------------------
__global__ __launch_bounds__(128) void edge_gemm(
    const float* __restrict__ Abuf, const float* __restrict__ edgeW,
    const int* __restrict__ tileEdge, float* __restrict__ Cbuf) {
  const int tile = blockIdx.x;
  const int e = tileEdge[tile];
  if (e < 0) return;                 // uniform across block
  __shared__ float As[32 * LDA];
  const int t = threadIdx.x;
  {
    const int r = t >> 2, q = t & 3;
    const float4* src = (const float4*)(Abuf + ((size_t)tile * 32 + r) * DD) + q * 8;
    float4* dst = (float4*)(As + r * LDA) + q * 8;
    #pragma unroll
    for (int i = 0; i < 8; ++i) dst[i] = src[i];
  }
  __syncthreads();
  const int wv = t >> 5, lane = t & 31;
  v8f c00 = {0.f,0.f,0.f,0.f,0.f,0.f,0.f,0.f}, c01 = c00, c10 = c00, c11 = c00;
  wmma_32x128(As, edgeW + (size_t)e * DD * DD, c00, c01, c10, c11, wv, lane);
  const int half = lane >> 4, lr = lane & 15;
  const int col = wv * 32 + lr;
  #pragma unroll
  for (int v = 0; v < 8; ++v) {
    const int r = v + half * 8;
    size_t o0 = ((size_t)tile * 32 + r) * DD;
    size_t o1 = ((size_t)tile * 32 + 16 + r) * DD;
    Cbuf[o0 + col]      = c00[v];
    Cbuf[o0 + col + 16] = c01[v];
    Cbuf[o1 + col]      = c10[v];
    Cbuf[o1 + col + 16] = c11[v];
  }
}

// ------------------------------------------------------------------
// Combine: h = (raw[p] + C[child0] + C[child1] + edge_b[e0] + edge_b[e1]) / 3
// ret[p] = relu(h); scores[p] = dot(h, score_W). One wave32 per parent.
// ------------------------------------------------------------------
__global__ __launch_bounds__(256) void level_combine(
    const float* __restrict__ raw, const float* __restrict__ Cbuf,
    const int* __restrict__ pos, const int* __restrict__ edges,
    const float* __restrict__ edge_b, const float* __restrict__ score_W,
    float* __restrict__ ret, float* __restrict__ scores,
    int p0, int c0, int npar) {
  const int wv = threadIdx.x >> 5, lane = threadIdx.x & 31;
  const int jp = blockIdx.x * 8 + wv;
  if (jp >= npar) return;
  const int p = p0 + jp;
  const int s0 = pos[2 * jp], s1 = pos[2 * jp + 1];
  const int e0 = edges[c0 + 2 * jp], e1 = edges[c0 + 2 * jp + 1];
  const float* r0 = Cbuf + (size_t)s0 * DD;
  const float* r1 = Cbuf + (size_t)s1 * DD;
  const float* rw = raw + (size_t)p * DD;
  float part = 0.f;
  #pragma unroll
  for (int i = 0; i < 4; ++i) {
    int col = lane + i * 32;
    float h = (rw[col] + r0[col] + r1[col]
               + edge_b[e0 * DD + col] + edge_b[e1 * DD + col]) / 3.0f;
    ret[(size_t)p * DD + col] = fmaxf(h, 0.f);
    part += h * score_W[col];
  }
  for (int off = 16; off > 0; off >>= 1) part += __shfl_down(part, off, 32);
  if (lane == 0) scores[p] = part;
}

// leaves: emb == raw
__global__ __launch_bounds__(256) void leaf_scores(
    const float* __restrict__ raw, const float* __restrict__ score_W,
    float* __restrict__ scores) {
  const int wv = threadIdx.x >> 5, lane = threadIdx.x & 31;
  const int leaf = LEAF0 + blockIdx.x * 8 + wv;
  if (leaf >= NN) return;
  const float* rw = raw + (size_t)leaf * DD;
  float part = 0.f;
  #pragma unroll
  for (int i = 0; i < 4; ++i) {
    int col = lane + i * 32;
    part += rw[col] * score_W[col];
  }
  for (int off = 16; off > 0; off >>= 1) part += __shfl_down(part, off, 32);
  if (lane == 0) scores[leaf] = part;
}

// ------------------------------------------------------------------
extern "C" void kernel_launch(void* const* d_in, const int* in_sizes, int n_in,
                              void* d_out, int out_size, void* d_ws, size_t ws_size,
                              hipStream_t stream) {
  const int*   data      = (const int*)d_in[0];
  const int*   edges     = (const int*)d_in[1];
  const float* data_vecs = (const float*)d_in[2];
  const float* data_W    = (const float*)d_in[3];
  const float* data_b    = (const float*)d_in[4];
  const float* edge_W    = (const float*)d_in[5];
  const float* edge_b    = (const float*)d_in[6];
  const float* score_W   = (const float*)d_in[7];
  float* scores = (float*)d_out;

  // workspace layout
  float* raw  = (float*)d_ws;                       // N*128
  float* ret  = raw  + (size_t)NN * DD;             // N*128
  float* Abuf = ret  + (size_t)NN * DD;             // MAXR*128
  float* Cbuf = Abuf + (size_t)MAXR * DD;           // MAXR*128
  int*   pos      = (int*)(Cbuf + (size_t)MAXR * DD); // NLEAF ints
  int*   tileEdge = pos + NLEAF;                      // up to 8192 ints
  int*   cnt      = tileEdge + 8192;
  int*   offs     = cnt + EE;
  int*   cursor   = offs + EE;

  // 1) raw for all nodes (+ ret for leaves)
  init_gemm<<<(NN + 31) / 32, 128, 0, stream>>>(data, data_vecs, data_W, data_b, raw, ret);

  // 2) leaf scores (emb == raw at leaves)
  leaf_scores<<<NLEAF / 8, 256, 0, stream>>>(raw, score_W, scores);

  // 3) bottom-up levels
  for (int l = LTREE - 2; l >= 0; --l) {
    const int p0 = (1 << l) - 1,       npar = 1 << l;
    const int c0 = (1 << (l + 1)) - 1, M    = 1 << (l + 1);
    const int ntiles = (M + 8 * (ALIGNR - 1) + (ALIGNR - 1)) / ALIGNR; // upper bound

    meta_init    <<<1, 32, 0, stream>>>(cnt, cursor);
    level_hist   <<<(M + 255) / 256, 256, 0, stream>>>(edges, c0, M, cnt);
    level_offsets<<<1, 256, 0, stream>>>(cnt, offs, tileEdge, ntiles, Abuf);
    level_gather <<<(M + 63) / 64, 256, 0, stream>>>(edges, ret, c0, M, offs, cursor, pos, Abuf);
    edge_gemm    <<<ntiles, 128, 0, stream>>>(Abuf, edge_W, tileEdge, Cbuf);
    level_combine<<<(npar + 7) / 8, 256, 0, stream>>>(raw, Cbuf, pos, edges, edge_b,
                                                      score_W, ret, scores, p0, c0, npar);
  }
}